// SoftDTW_20675972563827
// MI455X (gfx1250) — compile-verified
//
#include <hip/hip_runtime.h>
#include <math.h>

// Soft-DTW for MI455X (gfx1250, wave32).
// One workgroup (224 threads = 7 wave32s) per batch b in [0,256).
// Anti-diagonal wavefront DP in LDS with 3 rotating diagonal buffers.
// CDNA5-specific paths: global_load_async_to_lds_b32 (ASYNCcnt) staging,
// s_wait_asynccnt, wave32-sized workgroups.
// Critical-path optimization: R[i][j-1] (own previous value) is kept in a
// register, so each step's dependent chain is barrier -> 2 LDS loads ->
// exp/log chain -> LDS store instead of 3 LDS loads.

#define GAMMA_F   0.01f
#define INVG_F    100.0f          // 1/GAMMA
#define BIG_F     1.0e10f
#define L_        205             // ceil(2048/10) subsampled timesteps; N = M = L
#define BATCH_    256             // 128*2 features -> batch dim of the DP
#define FSTRIDE_  256             // floats per timestep in x/y (128*2)
#define GSTRIDE_  (10 * FSTRIDE_) // element stride between subsampled rows = 2560
#define DIAGN_    (L_ + 3)        // 208 entries per diagonal buffer
#define TPB_      224             // 7 wave32s, smallest multiple-of-32 >= 205

__global__ __launch_bounds__(TPB_) void SoftDTW_20675972563827_dp(
    const float* __restrict__ x, const float* __restrict__ y,
    float* __restrict__ partial)
{
    __shared__ float xsh[L_];
    __shared__ float ysh[L_];
    __shared__ float dbuf[3][DIAGN_];

    const int t = threadIdx.x;
    const int b = blockIdx.x;

    // ---- Stage the two strided columns X[b,:], Y[b,:] into LDS via the
    //      CDNA5 async global->LDS path (tracked by ASYNCcnt). Each active
    //      lane issues a per-lane 4B async copy from vaddr to the LDS offset in vdst. ----
    if (t < L_) {
        const float* gx = x + (size_t)t * GSTRIDE_ + b;
        const float* gy = y + (size_t)t * GSTRIDE_ + b;
        unsigned lx = (unsigned)(size_t)&xsh[t];   // low 32 bits = LDS offset
        unsigned ly = (unsigned)(size_t)&ysh[t];
        asm volatile("global_load_async_to_lds_b32 %0, %1, off"
                     :: "v"(lx), "v"(gx) : "memory");
        asm volatile("global_load_async_to_lds_b32 %0, %1, off"
                     :: "v"(ly), "v"(gy) : "memory");
    }

    // Init all three diagonal buffers to BIG while the async copies fly.
    for (int k = t; k < 3 * DIAGN_; k += TPB_) (&dbuf[0][0])[k] = BIG_F;

    asm volatile("s_wait_asynccnt 0" ::: "memory");  // this wave's LDS writes done
    __syncthreads();                                  // all waves' writes visible
    if (t == 0) dbuf[0][0] = 0.0f;                    // R[0][0] = 0
    __syncthreads();

    float* d2 = dbuf[0];   // diagonal i+j = p      (R[i-1][j-1])
    float* d1 = dbuf[1];   // diagonal i+j = p+1    (R[i-1][j])
    float* d0 = dbuf[2];   // diagonal i+j = p+2    (write target)

    const int   i    = t + 1;                     // this thread's row, 1..205
    const float xi   = (t < L_) ? xsh[t] : 0.0f;  // X[b, i-1]
    float       mine = BIG_F;                     // R[i][j-1]: own value, last step
                                                  // (init BIG == R[i][0] boundary)

    for (int p = 0; p < 2 * L_ - 1; ++p) {
        if (t < L_) {
            const int j = p + 2 - i;
            if (j >= 1 && j <= L_) {
                const float r0 = -d2[i - 1] * INVG_F;   // R[i-1][j-1] (LDS)
                const float r1 = -d1[i - 1] * INVG_F;   // R[i-1][j]   (LDS)
                const float r2 = -mine      * INVG_F;   // R[i][j-1]   (register)
                const float rmax = fmaxf(fmaxf(r0, r1), r2);
                const float rsum = __expf(r0 - rmax) + __expf(r1 - rmax)
                                 + __expf(r2 - rmax);
                const float softmin = -GAMMA_F * (__logf(rsum) + rmax);
                const float diff = xi - ysh[j - 1];     // D computed inline
                mine = diff * diff + softmin;
                d0[i] = mine;                           // published for row i+1
            }
        }
        if (t == 0) d0[0] = BIG_F;   // R[0][j>=1] boundary for future reads
        __syncthreads();
        float* tp = d2; d2 = d1; d1 = d0; d0 = tp;     // rotate diagonals
    }

    // Thread owning row 205 computed R[N][M] at p = 408; it is in `mine`.
    if (t == L_ - 1) partial[b] = mine;
}

// Deterministic fixed-tree reduction of the 256 per-batch partials.
__global__ __launch_bounds__(256) void SoftDTW_20675972563827_reduce(
    const float* __restrict__ partial, float* __restrict__ out)
{
    __shared__ float s[BATCH_];
    const int t = threadIdx.x;
    s[t] = partial[t];
    __syncthreads();
    for (int off = BATCH_ / 2; off > 0; off >>= 1) {
        if (t < off) s[t] += s[t + off];
        __syncthreads();
    }
    if (t == 0) out[0] = s[0];
}

extern "C" void kernel_launch(void* const* d_in, const int* in_sizes, int n_in,
                              void* d_out, int out_size, void* d_ws, size_t ws_size,
                              hipStream_t stream) {
    (void)in_sizes; (void)n_in; (void)out_size; (void)ws_size;
    const float* x = (const float*)d_in[0];   // (2048,128,2) float32
    const float* y = (const float*)d_in[1];   // (2048,128,2) float32
    float* partial = (float*)d_ws;            // 256 floats of scratch
    float* out     = (float*)d_out;           // 1 float

    SoftDTW_20675972563827_dp<<<BATCH_, TPB_, 0, stream>>>(x, y, partial);
    SoftDTW_20675972563827_reduce<<<1, 256, 0, stream>>>(partial, out);
}